// AutoEncoder_20822001451040
// MI455X (gfx1250) — compile-verified
//
#include <hip/hip_runtime.h>

// Problem constants (match reference)
#define NN 50000
#define EE 800000
#define BN_EPS 1e-5f

typedef float v2f __attribute__((ext_vector_type(2)));
typedef float v8f __attribute__((ext_vector_type(8)));

static inline int divUp(int a, int b) { return (a + b - 1) / b; }

// ---------------------------------------------------------------------------
// GEMM: H[N x FO] = X[N x K] @ W[K x FO], fp32 WMMA 16x16x4.
// Each wave computes a 16-row strip across ALL FO columns (FO/16 accumulators),
// so the A fragment is loaded once per k-step and reused for every col tile.
// Block = 256 threads = 8 waves = 128 rows. N = 50000 = 3125 * 16 row tiles.
// ---------------------------------------------------------------------------
template <int K, int FO>
__global__ __launch_bounds__(256) void gemm_wmma(const float* __restrict__ X,
                                                 const float* __restrict__ W,
                                                 float* __restrict__ H) {
  constexpr int ROW_TILES = NN / 16;  // 3125, exact
  constexpr int CT = FO / 16;

  const int lane = threadIdx.x & 31;
  const int wave = threadIdx.x >> 5;
  const int rowTile = blockIdx.x * 8 + wave;
  if (rowTile >= ROW_TILES) return;  // wave-uniform: EXEC stays all-ones

  const int row0 = rowTile * 16;
  const int half = lane >> 4;  // 0: lanes 0-15, 1: lanes 16-31
  const int l15 = lane & 15;

  v8f acc[CT];
  {
    v8f z = {};
    for (int i = 0; i < CT; ++i) acc[i] = z;
  }

  const float* xrow = X + (size_t)(row0 + l15) * K;

  for (int k0 = 0; k0 < K; k0 += 4) {
    // A 16x4 fp32 fragment: VGPR0 holds K = k0 + 2*half, VGPR1 = +1
    v2f a;
    a.x = xrow[k0 + 2 * half + 0];
    a.y = xrow[k0 + 2 * half + 1];
    const float* wk0 = W + (size_t)(k0 + 2 * half) * FO;
    const float* wk1 = W + (size_t)(k0 + 2 * half + 1) * FO;
#pragma unroll
    for (int ct = 0; ct < CT; ++ct) {
      const int col = ct * 16 + l15;
      v2f b;
      b.x = wk0[col];
      b.y = wk1[col];
      // 8 args: (neg_a, A, neg_b, B, c_mod, C, reuse_a, reuse_b)
      acc[ct] = __builtin_amdgcn_wmma_f32_16x16x4_f32(
          false, a, false, b, (short)0, acc[ct], false, false);
    }
  }

  // C/D layout: VGPR v holds row M = v (lanes 0-15) / M = v+8 (lanes 16-31)
#pragma unroll
  for (int ct = 0; ct < CT; ++ct) {
    const int col = ct * 16 + l15;
#pragma unroll
    for (int v = 0; v < 8; ++v) {
      H[(size_t)(row0 + v + 8 * half) * FO + col] = acc[ct][v];
    }
  }
}

// ---------------------------------------------------------------------------
// Degree / normalization precompute
// ---------------------------------------------------------------------------
__global__ void k_deg(const long long* __restrict__ dst, float* __restrict__ deg) {
  int e = blockIdx.x * blockDim.x + threadIdx.x;
  if (e >= EE) return;
  unsafeAtomicAdd(&deg[(int)dst[e]], 1.0f);
}

__global__ void k_node(const float* __restrict__ deg, float* __restrict__ dinv,
                       float* __restrict__ invd) {
  int i = blockIdx.x * blockDim.x + threadIdx.x;
  if (i >= NN) return;
  float d = deg[i] + 1.0f;  // self-loop
  dinv[i] = rsqrtf(d);
  invd[i] = 1.0f / d;
}

__global__ void k_enorm(const long long* __restrict__ src,
                        const long long* __restrict__ dst,
                        const float* __restrict__ dinv,
                        float* __restrict__ enorm) {
  int e = blockIdx.x * blockDim.x + threadIdx.x;
  if (e >= EE) return;
  enorm[e] = dinv[(int)src[e]] * dinv[(int)dst[e]];
}

// ---------------------------------------------------------------------------
// Edge gather-scale-scatter: AGG[dst] += H[src] * enorm  (float4 per thread)
// ---------------------------------------------------------------------------
template <int F>
__global__ __launch_bounds__(256) void edge_kernel(
    const long long* __restrict__ src, const long long* __restrict__ dst,
    const float* __restrict__ enorm, const float* __restrict__ H,
    float* __restrict__ AGG) {
  constexpr int C4 = F / 4;
  int tid = blockIdx.x * blockDim.x + threadIdx.x;
  int e = tid / C4;
  int c = tid % C4;
  if (e >= EE) return;
  long long s = src[e];
  long long d = dst[e];
  float nm = enorm[e];
  float4 h = ((const float4*)(H + (size_t)s * F))[c];
  float* a = AGG + (size_t)d * F + (size_t)c * 4;
  unsafeAtomicAdd(a + 0, h.x * nm);
  unsafeAtomicAdd(a + 1, h.y * nm);
  unsafeAtomicAdd(a + 2, h.z * nm);
  unsafeAtomicAdd(a + 3, h.w * nm);
}

// ---------------------------------------------------------------------------
// Fused: Y = AGG + H*inv_deg + b (written in place into AGG) + BN statistics
// Per-feature sum / sumsq reduced in LDS, one atomic per feature per block.
// ---------------------------------------------------------------------------
template <int F>
__global__ __launch_bounds__(256) void stats_kernel(
    float* __restrict__ Y, const float* __restrict__ H,
    const float* __restrict__ invd, const float* __restrict__ b,
    float* __restrict__ gsum, float* __restrict__ gsq) {
  constexpr int RPB = 256 / F;  // row sub-lanes per block
  const int f = threadIdx.x % F;
  const int rsub = threadIdx.x / F;
  const float bf = b[f];

  float s = 0.0f, sq = 0.0f;
  for (int r = blockIdx.x * RPB + rsub; r < NN; r += gridDim.x * RPB) {
    size_t idx = (size_t)r * F + f;
    float y = Y[idx] + H[idx] * invd[r] + bf;
    Y[idx] = y;
    s += y;
    sq += y * y;
  }

  __shared__ float ssum[256];
  __shared__ float ssq[256];
  ssum[threadIdx.x] = s;
  ssq[threadIdx.x] = sq;
  __syncthreads();
  if (rsub == 0) {
#pragma unroll
    for (int i = 1; i < RPB; ++i) {
      s += ssum[i * F + f];
      sq += ssq[i * F + f];
    }
    unsafeAtomicAdd(&gsum[f], s);
    unsafeAtomicAdd(&gsq[f], sq);
  }
}

// ---------------------------------------------------------------------------
// BN apply (+ optional ReLU): out = (y - m) * rsqrt(v + eps) * g + be
// ---------------------------------------------------------------------------
template <int F>
__global__ __launch_bounds__(256) void bn_apply(
    const float* __restrict__ Y, const float* __restrict__ gsum,
    const float* __restrict__ gsq, const float* __restrict__ g,
    const float* __restrict__ be, float* __restrict__ out, int relu) {
  int tid = blockIdx.x * blockDim.x + threadIdx.x;
  if (tid >= NN * F) return;
  int f = tid % F;
  const float invN = 1.0f / (float)NN;
  float m = gsum[f] * invN;
  float v = gsq[f] * invN - m * m;
  float y = (Y[tid] - m) * rsqrtf(v + BN_EPS) * g[f] + be[f];
  if (relu) y = fmaxf(y, 0.0f);
  out[tid] = y;
}

// ---------------------------------------------------------------------------
// One full GCN+BN(+ReLU) layer
// ---------------------------------------------------------------------------
template <int K, int FO>
static void run_layer(const float* xin, const float* W, const float* b,
                      const float* g, const float* be, const long long* src,
                      const long long* dst, const float* enorm,
                      const float* invd, float* H, float* AGG, float* gsum,
                      float* gsq, float* out, int relu, hipStream_t stream) {
  gemm_wmma<K, FO><<<divUp(NN / 16, 8), 256, 0, stream>>>(xin, W, H);
  hipMemsetAsync(AGG, 0, (size_t)NN * FO * sizeof(float), stream);
  hipMemsetAsync(gsum, 0, 2 * 128 * sizeof(float), stream);  // gsum+gsq contiguous
  edge_kernel<FO><<<divUp(EE * (FO / 4), 256), 256, 0, stream>>>(src, dst, enorm,
                                                                 H, AGG);
  stats_kernel<FO><<<512, 256, 0, stream>>>(AGG, H, invd, b, gsum, gsq);
  bn_apply<FO><<<divUp(NN * FO, 256), 256, 0, stream>>>(AGG, gsum, gsq, g, be,
                                                        out, relu);
}

extern "C" void kernel_launch(void* const* d_in, const int* in_sizes, int n_in,
                              void* d_out, int out_size, void* d_ws,
                              size_t ws_size, hipStream_t stream) {
  const float* x = (const float*)d_in[0];
  const long long* ei = (const long long*)d_in[1];  // int64 (2, E)
  const long long* src = ei;
  const long long* dst = ei + EE;

  // Per-layer params: inputs are x, edge_index, then (W,b,g,be) x 6
  const float* W[6];
  const float* b[6];
  const float* g[6];
  const float* be[6];
  for (int j = 0; j < 6; ++j) {
    W[j] = (const float*)d_in[2 + 4 * j];
    b[j] = (const float*)d_in[3 + 4 * j];
    g[j] = (const float*)d_in[4 + 4 * j];
    be[j] = (const float*)d_in[5 + 4 * j];
  }

  // Workspace layout (floats)
  float* ws = (float*)d_ws;
  float* XA = ws;                            // N*128 ping buffer
  float* H = XA + (size_t)NN * 128;          // N*128
  float* AGG = H + (size_t)NN * 128;         // N*128
  float* deg = AGG + (size_t)NN * 128;       // N
  float* dinv = deg + NN;                    // N
  float* invd = dinv + NN;                   // N
  float* enorm = invd + NN;                  // E
  float* gsum = enorm + EE;                  // 128
  float* gsq = gsum + 128;                   // 128

  // Graph normalization precompute
  hipMemsetAsync(deg, 0, NN * sizeof(float), stream);
  k_deg<<<divUp(EE, 256), 256, 0, stream>>>(dst, deg);
  k_node<<<divUp(NN, 256), 256, 0, stream>>>(deg, dinv, invd);
  k_enorm<<<divUp(EE, 256), 256, 0, stream>>>(src, dst, dinv, enorm);

  // Encoder: 128->128 (relu), 128->128 (relu), 128->64 (no relu)
  run_layer<128, 128>(x, W[0], b[0], g[0], be[0], src, dst, enorm, invd, H, AGG,
                      gsum, gsq, XA, 1, stream);
  run_layer<128, 128>(XA, W[1], b[1], g[1], be[1], src, dst, enorm, invd, H,
                      AGG, gsum, gsq, XA, 1, stream);
  run_layer<128, 64>(XA, W[2], b[2], g[2], be[2], src, dst, enorm, invd, H, AGG,
                     gsum, gsq, XA, 0, stream);
  // Decoder: 64->128 (relu), 128->128 (relu), 128->128 (no relu) -> d_out
  run_layer<64, 128>(XA, W[3], b[3], g[3], be[3], src, dst, enorm, invd, H, AGG,
                     gsum, gsq, XA, 1, stream);
  run_layer<128, 128>(XA, W[4], b[4], g[4], be[4], src, dst, enorm, invd, H,
                      AGG, gsum, gsq, XA, 1, stream);
  run_layer<128, 128>(XA, W[5], b[5], g[5], be[5], src, dst, enorm, invd, H,
                      AGG, gsum, gsq, (float*)d_out, 0, stream);
}